// GCNEncoder_8289286881626
// MI455X (gfx1250) — compile-verified
//
#include <hip/hip_runtime.h>

#define NN 10000
#define NE 640000
#define KDIM 128   // inner dim for both GEMMs (IN_CH == HID == 128)

typedef __attribute__((ext_vector_type(2))) float v2f;
typedef __attribute__((ext_vector_type(8))) float v8f;

// ---------------------------------------------------------------- utilities
__global__ void fill_kernel(float* __restrict__ p, long long n, float v) {
  long long i = (long long)blockIdx.x * blockDim.x + threadIdx.x;
  if (i < n) p[i] = v;
}

// deg[dst] += w[e]   (deg pre-filled with 1.0 to account for self-loop)
__global__ void degree_kernel(const long long* __restrict__ ei,
                              const float* __restrict__ w,
                              float* __restrict__ deg) {
  int e = blockIdx.x * blockDim.x + threadIdx.x;
  if (e < NE) {
    int dst = (int)ei[NE + e];
    atomicAdd(&deg[dst], w[e]);
  }
}

// in-place: deg[i] -> rsqrt(deg[i])
__global__ void dinv_kernel(float* __restrict__ deg) {
  int i = blockIdx.x * blockDim.x + threadIdx.x;
  if (i < NN) {
    float d = deg[i];
    deg[i] = (d > 0.0f) ? rsqrtf(d) : 0.0f;
  }
}

// norm[e] = dinv[src] * w[e] * dinv[dst]  (shared by both layers)
__global__ void norm_kernel(const long long* __restrict__ ei,
                            const float* __restrict__ w,
                            const float* __restrict__ dinv,
                            float* __restrict__ norm) {
  int e = blockIdx.x * blockDim.x + threadIdx.x;
  if (e < NE) {
    int s = (int)ei[e];
    int d = (int)ei[NE + e];
    norm[e] = dinv[s] * w[e] * dinv[d];
  }
}

// ------------------------------------------------------------- WMMA GEMM
// Y[NN, NCOL] = X[NN, KDIM] @ W[KDIM, NCOL], all f32 row-major.
// One wave per 16x16 output tile via V_WMMA_F32_16X16X4_F32.
// blockDim.x = 32 * (NCOL/16); gridDim.x = NN/16 (= 625, exact).
template <int NCOL>
__global__ void gemm_wmma_kernel(const float* __restrict__ X,
                                 const float* __restrict__ W,
                                 float* __restrict__ Y) {
  const int lane = threadIdx.x & 31;
  const int wave = threadIdx.x >> 5;
  const int m0 = blockIdx.x * 16;
  const int n0 = wave * 16;
  const int r  = lane & 15;          // row (A) / col (B,C,D) within tile
  const int kb = (lane >> 4) << 1;   // 0 for lanes 0-15, 2 for lanes 16-31

  v8f acc = {};
  const float* xrow = X + (long long)(m0 + r) * KDIM;
  const float* wcol = W + (n0 + r);

  for (int k = 0; k < KDIM; k += 4) {
    // A 16x4 f32 fragment: VGPR0 = K=kb, VGPR1 = K=kb+1 (ISA 7.12.2)
    v2f a, b;
    a.x = xrow[k + kb];
    a.y = xrow[k + kb + 1];
    // B 4x16 f32 fragment: lane holds column n0+r; VGPR pair holds K=kb,kb+1
    b.x = wcol[(long long)(k + kb) * NCOL];
    b.y = wcol[(long long)(k + kb + 1) * NCOL];
    acc = __builtin_amdgcn_wmma_f32_16x16x4_f32(
        /*neg_a=*/false, a, /*neg_b=*/false, b,
        /*c_mod=*/(short)0, acc, /*reuse_a=*/false, /*reuse_b=*/false);
  }

  // C/D layout: VGPR i -> M = i (lanes 0-15) or i+8 (lanes 16-31), N = r
  const int row_off = (lane >> 4) << 3;
#pragma unroll
  for (int i = 0; i < 8; ++i) {
    Y[(long long)(m0 + i + row_off) * NCOL + (n0 + r)] = acc[i];
  }
}

// ------------------------------------------------------------- edge scatter
// agg[dst, c] += norm[e] * xw[src, c]; one thread = one edge x 4 channels.
template <int C>
__global__ void scatter_kernel(const long long* __restrict__ ei,
                               const float* __restrict__ norm,
                               const float* __restrict__ xw,
                               float* __restrict__ agg) {
  constexpr int SH = (C == 128) ? 5 : 4;  // threads-per-edge = C/4 = 1<<SH
  long long idx = (long long)blockIdx.x * blockDim.x + threadIdx.x;
  int e = (int)(idx >> SH);
  if (e >= NE) return;
  int c4 = ((int)idx & ((1 << SH) - 1)) << 2;

  int s = (int)ei[e];
  int d = (int)ei[NE + e];
  float nw = norm[e];

  const float4 v = *(const float4*)(xw + (long long)s * C + c4);
  float* o = agg + (long long)d * C + c4;
  atomicAdd(o + 0, nw * v.x);
  atomicAdd(o + 1, nw * v.y);
  atomicAdd(o + 2, nw * v.z);
  atomicAdd(o + 3, nw * v.w);
}

// ------------------------------------------------------------- finalize
// out[i,c] = agg[i,c] + dinv[i]^2 * xw[i,c] + bias[c]  (optionally ReLU)
template <int C, bool RELU>
__global__ void finalize_kernel(const float* __restrict__ agg,
                                const float* __restrict__ xw,
                                const float* __restrict__ dinv,
                                const float* __restrict__ bias,
                                float* __restrict__ out) {
  constexpr int LOG = (C == 128) ? 7 : 6;
  long long idx = (long long)blockIdx.x * blockDim.x + threadIdx.x;
  if (idx >= (long long)NN * C) return;
  int i = (int)(idx >> LOG);
  int c = (int)idx & (C - 1);
  float di = dinv[i];
  float v = agg[idx] + di * di * xw[idx] + bias[c];
  if (RELU) v = fmaxf(v, 0.0f);
  out[idx] = v;
}

// ------------------------------------------------------------- launcher
extern "C" void kernel_launch(void* const* d_in, const int* in_sizes, int n_in,
                              void* d_out, int out_size, void* d_ws, size_t ws_size,
                              hipStream_t stream) {
  const float*     x  = (const float*)d_in[0];      // [10000,128]
  const long long* ei = (const long long*)d_in[1];  // [2,640000] int64
  const float*     ew = (const float*)d_in[2];      // [640000]
  const float*     W1 = (const float*)d_in[3];      // [128,128]
  const float*     b1 = (const float*)d_in[4];      // [128]
  const float*     W2 = (const float*)d_in[5];      // [128,64]
  const float*     b2 = (const float*)d_in[6];      // [64]
  float* out = (float*)d_out;                       // [10000,64]

  // workspace layout (floats); ~23 MB total
  float* dinv = (float*)d_ws;            // NN (deg -> dinv in place)
  float* norm = dinv + 10240;            // NE
  float* xw1  = norm + NE;               // NN*128
  float* agg1 = xw1 + (long long)NN * 128;
  float* h    = agg1 + (long long)NN * 128;
  float* xw2  = h + (long long)NN * 128;
  float* agg2 = xw2 + (long long)NN * 64;

  const int T = 256;
  auto blk = [](long long n, int t) { return (unsigned)((n + t - 1) / t); };

  // degrees (self-loop contributes weight 1.0) -> dinv -> per-edge norms
  fill_kernel<<<blk(NN, T), T, 0, stream>>>(dinv, NN, 1.0f);
  degree_kernel<<<blk(NE, T), T, 0, stream>>>(ei, ew, dinv);
  dinv_kernel<<<blk(NN, T), T, 0, stream>>>(dinv);
  norm_kernel<<<blk(NE, T), T, 0, stream>>>(ei, ew, dinv, norm);

  // ---- layer 1: xw1 = x @ W1 ; scatter ; + self-loop + bias ; ReLU -> h
  gemm_wmma_kernel<128><<<NN / 16, 8 * 32, 0, stream>>>(x, W1, xw1);
  fill_kernel<<<blk((long long)NN * 128, T), T, 0, stream>>>(agg1, (long long)NN * 128, 0.0f);
  scatter_kernel<128><<<blk((long long)NE * 32, T), T, 0, stream>>>(ei, norm, xw1, agg1);
  finalize_kernel<128, true><<<blk((long long)NN * 128, T), T, 0, stream>>>(agg1, xw1, dinv, b1, h);

  // ---- layer 2: xw2 = h @ W2 ; scatter ; + self-loop + bias -> out
  gemm_wmma_kernel<64><<<NN / 16, 4 * 32, 0, stream>>>(h, W2, xw2);
  fill_kernel<<<blk((long long)NN * 64, T), T, 0, stream>>>(agg2, (long long)NN * 64, 0.0f);
  scatter_kernel<64><<<blk((long long)NE * 16, T), T, 0, stream>>>(ei, norm, xw2, agg2);
  finalize_kernel<64, false><<<blk((long long)NN * 64, T), T, 0, stream>>>(agg2, xw2, dinv, b2, out);
}